// WeightedLoss_1992864825404
// MI455X (gfx1250) — compile-verified
//
#include <hip/hip_runtime.h>
#include <stdint.h>

#define BDIM   256
#define WAVES  (BDIM / 32)
#define CCOLS  4096
#define CHUNK  512                 // floats per staged chunk (2 KB)
#define NCHUNK (CCOLS / CHUNK)     // 8

// ---- CDNA5 async global->LDS path (probe-confirmed on this toolchain) ----
#if defined(__AMDGCN__) && __has_builtin(__builtin_amdgcn_global_load_async_to_lds_b128)
#define HAVE_ASYNC 1
#else
#define HAVE_ASYNC 0
#endif

#if __has_builtin(__builtin_amdgcn_s_wait_asynccnt)
#define WAIT_ASYNC(n) do { __builtin_amdgcn_s_wait_asynccnt(n); asm volatile("" ::: "memory"); } while (0)
#else
#define WAIT_ASYNC(n) asm volatile("s_wait_asynccnt %0" :: "n"(n) : "memory")
#endif

#if HAVE_ASYNC
// builtin signature: (v4i addrspace(1)*, v4i addrspace(3)*, imm offset, imm cpol)
typedef int v4i __attribute__((vector_size(16)));
typedef __attribute__((address_space(1))) v4i as1_v4i;
typedef __attribute__((address_space(3))) v4i as3_v4i;

__device__ __forceinline__ void async_copy_b128(const float* gsrc, float* ldst) {
  // per-lane 16B global->LDS DMA, tracked by ASYNCcnt; TH=NT (streamed once)
  __builtin_amdgcn_global_load_async_to_lds_b128((as1_v4i*)gsrc, (as3_v4i*)ldst,
                                                 /*imm offset*/0, /*cpol TH_NT*/1);
}
#endif

// Chunk-blocked online softmax over 16 in-register values per lane.
// Per-lane columns are ascending (col = cbase + (i>>2)*128 + (i&3)), so
// strict '>' keeps first-occurrence argmax; rescale happens once per chunk.
__device__ __forceinline__ void process16(const float x[16], int cbase,
                                          float& m, float& s, int& amax) {
  // pass 1: branchless chunk max + argmax
  float mc = x[0];
  int   ac = cbase;
  #pragma unroll
  for (int i = 1; i < 16; ++i) {
    int  col = cbase + ((i >> 2) * 128) + (i & 3);
    bool g   = x[i] > mc;
    mc = g ? x[i] : mc;
    ac = g ? col  : ac;
  }
  // merge into running stats: single rescale per chunk (exp(0)==1 when !g)
  bool  g = mc > m;
  float M = fmaxf(m, mc);
  s *= __expf(m - M);
  amax = g ? ac : amax;
  m = M;
  // pass 2: fma + native exp2 + add per element (exp is TRANS, co-executes)
  const float LOG2E = 1.44269504088896340736f;
  float ml2 = m * LOG2E;
  #pragma unroll
  for (int i = 0; i < 16; ++i)
    s += exp2f(fmaf(x[i], LOG2E, -ml2));
}

__global__ __launch_bounds__(BDIM) void wloss_rows(
    const float* __restrict__ outp, const int* __restrict__ tgt,
    const float* __restrict__ wmat, float* __restrict__ rowloss, int nrows) {
#if HAVE_ASYNC
  __shared__ float smem[2][WAVES][CHUNK];   // 32 KB: per-wave double buffer
#endif
  const int lane = threadIdx.x & 31;
  const int wv   = threadIdx.x >> 5;
  const int row  = blockIdx.x * WAVES + wv;
  if (row >= nrows) return;                 // wave-uniform; no barriers used

  const float* rowp = outp + (size_t)row * CCOLS;

  float m = -__builtin_inff();
  float s = 0.0f;
  int   amax = 0;

#if HAVE_ASYNC
  // prime chunk 0
  {
    float* dst = &smem[0][wv][0];
    #pragma unroll
    for (int j = 0; j < 4; ++j)
      async_copy_b128(rowp + j * 128 + lane * 4, dst + j * 128 + lane * 4);
  }
  for (int c = 0; c < NCHUNK; ++c) {
    if (c + 1 < NCHUNK) {
      const float* src = rowp + (c + 1) * CHUNK;
      float*       dst = &smem[(c + 1) & 1][wv][0];
      #pragma unroll
      for (int j = 0; j < 4; ++j)
        async_copy_b128(src + j * 128 + lane * 4, dst + j * 128 + lane * 4);
      WAIT_ASYNC(4);                        // chunk c done; c+1 still in flight
    } else {
      WAIT_ASYNC(0);                        // drain last chunk
    }
    const float* buf = &smem[c & 1][wv][0];
    float x[16];
    #pragma unroll
    for (int j = 0; j < 4; ++j) {
      float4 v = *reinterpret_cast<const float4*>(buf + j * 128 + lane * 4);
      x[4 * j + 0] = v.x; x[4 * j + 1] = v.y;
      x[4 * j + 2] = v.z; x[4 * j + 3] = v.w;
    }
    process16(x, c * CHUNK + lane * 4, m, s, amax);
  }
#else
  // fallback: coalesced float4 stream with prefetch
  for (int c = 0; c < NCHUNK; ++c) {
    float x[16];
    #pragma unroll
    for (int j = 0; j < 4; ++j) {
      const float* p = rowp + c * CHUNK + j * 128 + lane * 4;
      __builtin_prefetch(p + CHUNK, 0, 0);  // speculative, OOB-safe
      float4 v = *reinterpret_cast<const float4*>(p);
      x[4 * j + 0] = v.x; x[4 * j + 1] = v.y;
      x[4 * j + 2] = v.z; x[4 * j + 3] = v.w;
    }
    process16(x, c * CHUNK + lane * 4, m, s, amax);
  }
#endif

  // wave32 reduction: combine (max, first-occurrence argmax, rescaled sumexp)
  #pragma unroll
  for (int off = 16; off > 0; off >>= 1) {
    float m2 = __shfl_xor(m, off, 32);
    float s2 = __shfl_xor(s, off, 32);
    int   a2 = __shfl_xor(amax, off, 32);
    float mn = fmaxf(m, m2);
    s = s * __expf(m - mn) + s2 * __expf(m2 - mn);
    if (m2 > m || (m2 == m && a2 < amax)) amax = a2;
    m = mn;
  }

  if (lane == 0) {
    int   t  = tgt[row];
    float xt = rowp[t];                       // single 4B gather per row
    float ce = (m + __logf(s)) - xt;          // -log_softmax at target
    float w  = wmat[(size_t)t * CCOLS + amax];
    rowloss[row] = ce * w;
  }
}

// deterministic mean: fixed-order strided f64 accumulation + LDS tree
__global__ __launch_bounds__(256) void reduce_mean(
    const float* __restrict__ v, float* __restrict__ outp, int n) {
  __shared__ double sh[256];
  double acc = 0.0;
  for (int i = threadIdx.x; i < n; i += 256) acc += (double)v[i];
  sh[threadIdx.x] = acc;
  __syncthreads();
  for (int sft = 128; sft > 0; sft >>= 1) {
    if ((int)threadIdx.x < sft) sh[threadIdx.x] += sh[threadIdx.x + sft];
    __syncthreads();
  }
  if (threadIdx.x == 0) outp[0] = (float)(sh[0] / (double)n);
}

extern "C" void kernel_launch(void* const* d_in, const int* in_sizes, int n_in,
                              void* d_out, int out_size, void* d_ws, size_t ws_size,
                              hipStream_t stream) {
  const float* outp = (const float*)d_in[0];   // [B, C] f32
  const int*   tgt  = (const int*)d_in[1];     // [B] int
  const float* wmat = (const float*)d_in[2];   // [C, C] f32
  const int B = in_sizes[1];                   // 32768

  float* rowloss = (float*)d_ws;               // B floats of scratch

  const int nblocks = (B + WAVES - 1) / WAVES; // one wave32 per row
  wloss_rows<<<nblocks, BDIM, 0, stream>>>(outp, tgt, wmat, rowloss, B);
  reduce_mean<<<1, 256, 0, stream>>>(rowloss, (float*)d_out, B);
}